// AlpamayoR1_28509992910956
// MI455X (gfx1250) — compile-verified
//
#include <hip/hip_runtime.h>

// ---------------------------------------------------------------------------
// Batched banded (half-bandwidth 3) SPD Cholesky solve for the spline-velocity
// system.  ATA is tridiagonal (cos^2+sin^2=1), DTD is the constant 7-banded
// third-difference Gram matrix, so lhs = band(3).  One thread = one system.
// Memory-bound (~8.5 MB total @ 23.3 TB/s): inputs staged to LDS with CDNA5
// async-to-LDS B128 ops, outputs streamed back with async-from-LDS.
// The constant lhs band table (DTD bands + ATA diag + ridge) is computed once
// per block, in parallel, overlapped with the async staging.
// ---------------------------------------------------------------------------

#define TPB   64          // threads (= systems) per block, 2 wave32 waves
#define NP    65          // theta length / output row length
#define NS    64          // unknowns per system
#define DXROW 132         // padded LDS row stride for dxy (128 data + 4 pad)
#define LAM3  1e-4f       // V_LAMBDA / DT^6
#define RIDGE 1e-4f

// stencil s = {-1, 3, -3, 1} of the third difference operator
__device__ __forceinline__ float s3v(int k) {
  float m = (k == 0 || k == 3) ? 1.f : 3.f;
  return (k & 1) ? m : -m;
}
// (D3^T D3)[i][j] for j <= i, i-j <= 3, Np=65 (D3 has M=62 rows)
__device__ __forceinline__ float gband(int i, int j) {
  float acc = 0.f;
  int mlo = i - 3; if (mlo < 0) mlo = 0;
  int mhi = j;     if (mhi > NP - 4) mhi = NP - 4;   // M-1 = 61
  for (int m = mlo; m <= mhi; ++m) acc += s3v(i - m) * s3v(j - m);
  return acc;
}
// fill the 4 lhs band entries for row p (global index i = p+1)
__device__ __forceinline__ void band_row(int p, float& bd, float& b1, float& b2, float& b3) {
  int i = p + 1;
  bd = ((p == NS - 1) ? 1.f : 2.f) + LAM3 * gband(i, i) + RIDGE;  // full diagonal
  b1 = LAM3 * gband(i, i - 1);   // add c[p] at runtime
  b2 = LAM3 * gband(i, i - 2);
  b3 = LAM3 * gband(i, i - 3);
}

// One banded Cholesky solve.  th/orow may alias (orow written only after all
// th reads are done).  Work arrays use compile-time stride WS (lane-interleaved
// in LDS for the block kernel, contiguous scratch for the tail kernel).
// bd/b1/b2/b3: precomputed constant band table, stride 1 (broadcast reads).
template <int WS>
__device__ void solve_one(const float* th, const float* __restrict__ dx, float v0,
                          const float* bd, const float* b1, const float* b2, const float* b3,
                          float* l0a, float* l1a, float* l2a, float* l3a, float* za,
                          float* orow)
{
  float ctp, stp;                       // cos/sin(theta[p]) sliding state
  __sincosf(th[0], &stp, &ctp);
  float z1 = 0.f, z2 = 0.f, z3 = 0.f;   // z[p-1..p-3]
  float l0m1 = 1.f, l0m2 = 1.f, l0m3 = 1.f;
  float l1m1 = 0.f, l1m2 = 0.f, l2m1 = 0.f;

  for (int p = 0; p < NS; ++p) {
    float sn, cn;                       // sin/cos(theta[p+1])
    __sincosf(th[p + 1], &sn, &cn);
    float cp = ctp * cn + stp * sn;     // c[p] = cos(theta[p]-theta[p+1]) = ATA[p+1][p]

    // rhs[p] = A[:,p+1]^T b  (b = 2*dxy interleaved), minus v0 couplings
    float r = cn * dx[2 * p] + sn * dx[2 * p + 1];
    if (p < NS - 1) r += cn * dx[2 * p + 2] + sn * dx[2 * p + 3];
    r *= 2.f;
    if      (p == 0) r -= (cp - 3.f * LAM3) * v0;   // ATA[1][0]=c0, DTD[1][0]=-3*lam
    else if (p == 1) r -= 3.f * LAM3 * v0;          // DTD[2][0]=+3*lam
    else if (p == 2) r += LAM3 * v0;                // DTD[3][0]=-1*lam

    // lhs band entries for row p: broadcast table + runtime c[p]
    float d  = bd[p];
    float a1 = cp + b1[p];
    float a2 = b2[p];
    float a3 = b3[p];

    // banded Cholesky row p (lower bw 3) + fused forward substitution
    float l3p = (p >= 3) ? a3 / l0m3 : 0.f;
    float l2p = (p >= 2) ? (a2 - l3p * l1m2) / l0m2 : 0.f;
    float l1p = (p >= 1) ? (a1 - l3p * l2m1 - l2p * l1m1) / l0m1 : 0.f;
    float l0p = sqrtf(d - l3p * l3p - l2p * l2p - l1p * l1p);
    float zp  = (r - l1p * z1 - l2p * z2 - l3p * z3) / l0p;

    l0a[p * WS] = l0p; l1a[p * WS] = l1p; l2a[p * WS] = l2p;
    l3a[p * WS] = l3p; za [p * WS] = zp;

    z3 = z2; z2 = z1; z1 = zp;
    l0m3 = l0m2; l0m2 = l0m1; l0m1 = l0p;
    l1m2 = l1m1; l1m1 = l1p; l2m1 = l2p;
    ctp = cn; stp = sn;
  }

  // back substitution:  L^T y = z
  float y1 = 0.f, y2 = 0.f, y3 = 0.f;
  for (int p = NS - 1; p >= 0; --p) {
    float acc = za[p * WS];
    if (p + 1 <= NS - 1) acc -= l1a[(p + 1) * WS] * y1;
    if (p + 2 <= NS - 1) acc -= l2a[(p + 2) * WS] * y2;
    if (p + 3 <= NS - 1) acc -= l3a[(p + 3) * WS] * y3;
    float yp = acc / l0a[p * WS];
    orow[p + 1] = yp;
    y3 = y2; y2 = y1; y1 = yp;
  }
  orow[0] = v0;
}

// ---- CDNA5 async-to-LDS helpers (ASYNCcnt-tracked, GV addressing) ---------
__device__ __forceinline__ void async_g2l_b128(unsigned lds_off, const void* gaddr) {
  asm volatile("global_load_async_to_lds_b128 %0, %1, off"
               :: "v"(lds_off), "v"(gaddr) : "memory");
}
__device__ __forceinline__ void async_l2g_b128(const void* gaddr, unsigned lds_off) {
  asm volatile("global_store_async_from_lds_b128 %0, %1, off"
               :: "v"(gaddr), "v"(lds_off) : "memory");
}
__device__ __forceinline__ void wait_async0() {
  asm volatile("s_wait_asynccnt 0x0" ::: "memory");
}

// LDS layout (floats):
//   [0, DXROW*TPB)            dxy rows, stride 132 (16B-aligned rows, <=2-way conflicts)
//   [.., + NP*TPB)            theta rows (stride 65, odd -> conflict-free);
//                             reused as output rows (v0 | y)
//   then 5 lane-interleaved work arrays of NS*TPB each: L0,L1,L2,L3,Z
//   then 4 constant band tables of NS each: BD,B1,B2,B3 (broadcast reads)
#define SMEM_FLOATS (DXROW * TPB + NP * TPB + 5 * NS * TPB + 4 * NS)

__global__ __launch_bounds__(TPB)
void solve_block_kernel(const float* __restrict__ dxy, const float* __restrict__ theta,
                        const float* __restrict__ v0, float* __restrict__ out)
{
  extern __shared__ float smem[];
  const int tid = threadIdx.x;
  const int blk = blockIdx.x;

  float* s_dxy = smem;
  float* s_th  = smem + DXROW * TPB;
  float* s_l0  = s_th + NP * TPB;
  float* s_l1  = s_l0 + NS * TPB;
  float* s_l2  = s_l1 + NS * TPB;
  float* s_l3  = s_l2 + NS * TPB;
  float* s_z   = s_l3 + NS * TPB;
  float* s_bd  = s_z  + NS * TPB;
  float* s_b1  = s_bd + NS;
  float* s_b2  = s_b1 + NS;
  float* s_b3  = s_b2 + NS;

  // ---- issue async staging (coalesced B128, ASYNCcnt-tracked) -------------
  // dxy: 64 rows * 128 floats = 2048 vec4 into padded rows
  const float* gd = dxy + (size_t)blk * (TPB * 128);
  for (int it = 0; it < 32; ++it) {
    int q = it * TPB + tid;                 // vec4 index
    int row = q >> 5, col = q & 31;         // 32 vec4 per source row
    async_g2l_b128((unsigned)(size_t)(&s_dxy[row * DXROW + col * 4]), gd + 4 * q);
  }
  // theta: 64*65 = 4160 floats = 1040 vec4, identity layout
  const float* gt = theta + (size_t)blk * (TPB * NP);
  for (int it = 0; it < 16; ++it) {
    int q = it * TPB + tid;
    async_g2l_b128((unsigned)(size_t)(&s_th[4 * q]), gt + 4 * q);
  }
  if (tid < 16) {                           // tail 16 vec4
    int q = 16 * TPB + tid;
    async_g2l_b128((unsigned)(size_t)(&s_th[4 * q]), gt + 4 * q);
  }

  // ---- overlap: build constant lhs band table (independent of staged data)
  {
    float bd, b1, b2, b3;
    band_row(tid, bd, b1, b2, b3);          // TPB == NS: one row per thread
    s_bd[tid] = bd; s_b1[tid] = b1; s_b2[tid] = b2; s_b3[tid] = b3;
  }

  wait_async0();
  __syncthreads();

  const float vv = v0[blk * TPB + tid];
  // output row reuses this thread's theta row (theta fully consumed before writes)
  solve_one<TPB>(&s_th[tid * NP], &s_dxy[tid * DXROW], vv,
                 s_bd, s_b1, s_b2, s_b3,
                 &s_l0[tid], &s_l1[tid], &s_l2[tid], &s_l3[tid], &s_z[tid],
                 &s_th[tid * NP]);
  __syncthreads();

  // ---- stream (v0 | y) rows back, coalesced async LDS->global -------------
  float* go = out + (size_t)blk * (TPB * NP);
  for (int it = 0; it < 16; ++it) {
    int q = it * TPB + tid;
    async_l2g_b128(go + 4 * q, (unsigned)(size_t)(&s_th[4 * q]));
  }
  if (tid < 16) {
    int q = 16 * TPB + tid;
    async_l2g_b128(go + 4 * q, (unsigned)(size_t)(&s_th[4 * q]));
  }
  wait_async0();
}

// Fallback for B not divisible by TPB (not hit for B=8192): plain per-thread
// version with scratch arrays.
__global__ void solve_tail_kernel(const float* __restrict__ dxy, const float* __restrict__ theta,
                                  const float* __restrict__ v0, float* __restrict__ out,
                                  int b0, int B)
{
  int b = b0 + blockIdx.x * blockDim.x + threadIdx.x;
  if (b >= B) return;
  float th[NP], dx[128];
  for (int j = 0; j < NP; ++j)  th[j] = theta[(size_t)b * NP + j];
  for (int j = 0; j < 128; ++j) dx[j] = dxy[(size_t)b * 128 + j];
  float bd[NS], b1[NS], b2[NS], b3[NS];
  for (int p = 0; p < NS; ++p) band_row(p, bd[p], b1[p], b2[p], b3[p]);
  float l0[NS], l1[NS], l2[NS], l3[NS], z[NS], orow[NP];
  solve_one<1>(th, dx, v0[b], bd, b1, b2, b3, l0, l1, l2, l3, z, orow);
  for (int j = 0; j < NP; ++j) out[(size_t)b * NP + j] = orow[j];
}

extern "C" void kernel_launch(void* const* d_in, const int* in_sizes, int n_in,
                              void* d_out, int out_size, void* d_ws, size_t ws_size,
                              hipStream_t stream) {
  const float* dxy   = (const float*)d_in[0];   // [B,64,2]
  const float* theta = (const float*)d_in[1];   // [B,65]
  const float* v0    = (const float*)d_in[2];   // [B]
  float* out = (float*)d_out;                   // [B,65]

  int B = in_sizes[0] / 128;
  int nfull = B / TPB;
  if (nfull > 0) {
    size_t shmem = (size_t)SMEM_FLOATS * sizeof(float);   // ~133 KB (< 320 KB WGP LDS)
    solve_block_kernel<<<nfull, TPB, shmem, stream>>>(dxy, theta, v0, out);
  }
  int rem = B - nfull * TPB;
  if (rem > 0) {
    solve_tail_kernel<<<(rem + 63) / 64, 64, 0, stream>>>(dxy, theta, v0, out, nfull * TPB, B);
  }
}